// SHGR_71579924955643
// MI455X (gfx1250) — compile-verified
//
#include <hip/hip_runtime.h>
#include <math.h>

// Problem constants (from setup_inputs)
#define B_Q   2048
#define N_S   100000
#define DIM   128
#define DT_   8
#define KNN   5
#define NSPLIT 64
#define CHUNK  ((N_S + NSPLIT - 1) / NSPLIT)   // 1563
#define WAVES  4                                // waves per block in main kernel
#define LDS_STRIDE 132                          // 128 + 4 pad -> conflict-free ds reads

typedef float v2f __attribute__((ext_vector_type(2)));
typedef float v4f __attribute__((ext_vector_type(4)));
typedef float v8f __attribute__((ext_vector_type(8)));

#define NEG_INF (-3.402823466e38f)

// CDNA5 async global->LDS path (ASYNCcnt-tracked, no VGPR round trip)
#if __has_builtin(__builtin_amdgcn_global_load_async_to_lds_b128)
#define USE_ASYNC 1
#else
#define USE_ASYNC 0
#endif

#if __has_builtin(__builtin_amdgcn_s_wait_asynccnt)
#define WAIT_ASYNC(n) __builtin_amdgcn_s_wait_asynccnt(n)
#else
#define WAIT_ASYNC(n) asm volatile("s_wait_asynccnt %0" ::"i"(n) : "memory")
#endif

// 128-bit payload type + address-space-qualified pointers for the async builtin
typedef int v4i_b128 __attribute__((vector_size(16)));
typedef v4i_b128 __attribute__((address_space(1)))* g_b128_t;  // global
typedef v4i_b128 __attribute__((address_space(3)))* l_b128_t;  // LDS

// Branchless insert of (v,j) into sorted-descending top-5 registers t0..t4 / i0..i4
#define TOP5_INSERT(v, j)                                                     \
  do {                                                                        \
    bool c0 = (v) > t0, c1 = (v) > t1, c2 = (v) > t2, c3 = (v) > t3,          \
         c4 = (v) > t4;                                                       \
    t4 = c3 ? t3 : (c4 ? (v) : t4); i4 = c3 ? i3 : (c4 ? (j) : i4);           \
    t3 = c2 ? t2 : (c3 ? (v) : t3); i3 = c2 ? i2 : (c3 ? (j) : i3);           \
    t2 = c1 ? t1 : (c2 ? (v) : t2); i2 = c1 ? i1 : (c2 ? (j) : i2);           \
    t1 = c0 ? t0 : (c1 ? (v) : t1); i1 = c0 ? i0 : (c1 ? (j) : i1);           \
    t0 = c0 ? (v) : t0;             i0 = c0 ? (j) : i0;                       \
  } while (0)

// ---------------------------------------------------------------------------
// Kernel 1: per-row inverse L2 norm (one wave32 per 128-float row)
// ---------------------------------------------------------------------------
__global__ void rownorm_inv(const float* __restrict__ src,
                            float* __restrict__ dst, int rows) {
  const int wave = threadIdx.x >> 5;
  const int lane = threadIdx.x & 31;
  const int row  = blockIdx.x * 4 + wave;
  if (row >= rows) return;
  v4f x = *(const v4f*)(src + (size_t)row * DIM + lane * 4);
  float ss = x.x * x.x + x.y * x.y + x.z * x.z + x.w * x.w;
  ss += __shfl_xor(ss, 16, 32);
  ss += __shfl_xor(ss, 8, 32);
  ss += __shfl_xor(ss, 4, 32);
  ss += __shfl_xor(ss, 2, 32);
  ss += __shfl_xor(ss, 1, 32);
  if (lane == 0) dst[row] = 1.0f / fmaxf(sqrtf(ss), 1e-12f);
}

// ---------------------------------------------------------------------------
// Tile staging: 16 rows x 128 f32 of supports -> LDS (row-padded to 132)
// ---------------------------------------------------------------------------
#if USE_ASYNC
__device__ __forceinline__ void stage_async(const float* __restrict__ S,
                                            float* dst, unsigned tileBase,
                                            int lane, unsigned flatLim) {
  const unsigned gbase = tileBase * (unsigned)DIM + (unsigned)lane * 4u;
#pragma unroll
  for (int it = 0; it < 16; ++it) {
    unsigned flat = gbase + (unsigned)(it * DIM);
    if (flat > flatLim) flat = flatLim;  // clamp partial last tile (16B aligned)
    __builtin_amdgcn_global_load_async_to_lds_b128(
        (g_b128_t)(uintptr_t)(S + flat),
        (l_b128_t)(uintptr_t)(dst + it * LDS_STRIDE + lane * 4), 0, 0);
  }
}
#else
__device__ __forceinline__ void stage_sync(const float* __restrict__ S,
                                           float* dst, unsigned tileBase,
                                           int lane, unsigned flatLim) {
  const unsigned gbase = tileBase * (unsigned)DIM + (unsigned)lane * 4u;
#pragma unroll
  for (int it = 0; it < 16; ++it) {
    unsigned flat = gbase + (unsigned)(it * DIM);
    if (flat > flatLim) flat = flatLim;
    v4f d = *(const v4f*)(S + flat);
    *(v4f*)(dst + it * LDS_STRIDE + lane * 4) = d;
  }
}
#endif

// ---------------------------------------------------------------------------
// Kernel 2: WMMA f32 GEMM (16 queries x chunk of supports, K=128) fused with
// per-lane streaming top-5.  Mapping: A = 16 supports (M), B = 16 queries (N),
// so D-tile lane l holds query l%16, VGPR v holds support v + 8*(l/16).
// Double-buffered async global->LDS staging.
// ---------------------------------------------------------------------------
__global__ void __launch_bounds__(32 * WAVES)
knn_gemm_topk(const float* __restrict__ Q, const float* __restrict__ S,
              const float* __restrict__ invQ, const float* __restrict__ invS,
              float* __restrict__ pVal, int* __restrict__ pIdx) {
  __shared__ float ldsA[WAVES][2][16 * LDS_STRIDE];  // per-wave double buffer
  __shared__ float mv[16][2 * WAVES][KNN];           // merge buffer: values
  __shared__ int   mi[16][2 * WAVES][KNN];           // merge buffer: indices

  const int tid  = threadIdx.x;
  const int wave = tid >> 5;
  const int lane = tid & 31;
  const int lh   = lane >> 4;   // lane half
  const int lm   = lane & 15;   // query within tile / support row within tile

  const int qbase  = blockIdx.x * 16;
  const int split  = blockIdx.y;
  const int cStart = split * CHUNK;
  const int cEnd   = min(N_S, cStart + CHUNK);
  const int nTiles = (cEnd - cStart + 15) >> 4;

  // Persistent B-matrix (query) fragments: 32 K-steps of 2 VGPRs each.
  v2f bf[32];
  {
    const float* qrow = Q + (size_t)(qbase + lm) * DIM + 2 * lh;
#pragma unroll
    for (int kk = 0; kk < 32; ++kk) bf[kk] = *(const v2f*)(qrow + 4 * kk);
  }
  const float invq = invQ[qbase + lm];

  float t0 = NEG_INF, t1 = NEG_INF, t2 = NEG_INF, t3 = NEG_INF, t4 = NEG_INF;
  int   i0 = 0, i1 = 0, i2 = 0, i3 = 0, i4 = 0;

  float* buf0 = ldsA[wave][0];
  float* buf1 = ldsA[wave][1];
  const unsigned flatLim = (unsigned)(N_S * DIM - 4);

#if USE_ASYNC
  // Prologue: stage this wave's first tile.
  stage_async(S, buf0, (unsigned)(cStart + wave * 16), lane, flatLim);
#endif

  int cur = 0;
  for (int t = wave; t < nTiles; t += WAVES) {
    const unsigned tileBase = (unsigned)(cStart + t * 16);
    float* bufC = cur ? buf1 : buf0;
    float* bufN = cur ? buf0 : buf1;

    // Warm L2 two tiles ahead of the async engine.
    if (t + 2 * WAVES < nTiles) {
      unsigned pf = (tileBase + 32u * WAVES) * DIM + (unsigned)lane * 64u;
      if (pf > flatLim) pf = flatLim;
      __builtin_prefetch(S + pf, 0, 1);
    }

#if USE_ASYNC
    // WAR: ensure prior ds reads of bufN are fully retired before async
    // writes land there (DScnt and ASYNCcnt are mutually unordered).
    asm volatile("s_wait_dscnt 0" ::: "memory");
    if (t + WAVES < nTiles) {
      stage_async(S, bufN, tileBase + 16u * WAVES, lane, flatLim);
      WAIT_ASYNC(16);  // retire current buffer's 16 ops (in-order completion)
    } else {
      WAIT_ASYNC(0);
    }
#else
    stage_sync(S, bufC, tileBase, lane, flatLim);
    (void)bufN;
#endif

    // 32 chained f32 WMMAs: D(16x16) += A(16x4 supports) x B(4x16 queries)
    v8f c = {0.f, 0.f, 0.f, 0.f, 0.f, 0.f, 0.f, 0.f};
    const float* ar = bufC + lm * LDS_STRIDE + 2 * lh;
#pragma unroll
    for (int kk = 0; kk < 32; ++kk) {
      v2f a = *(const v2f*)(ar + 4 * kk);
      c = __builtin_amdgcn_wmma_f32_16x16x4_f32(
          /*neg_a=*/false, a, /*neg_b=*/false, bf[kk],
          /*c_mod=*/(short)0, c, /*reuse_a=*/false, /*reuse_b=*/false);
    }

    // Fused streaming top-5 (per lane = per query half).
#pragma unroll
    for (int v = 0; v < 8; ++v) {
      const int sIdx = (int)tileBase + v + 8 * lh;
      const int sClamp = sIdx < N_S ? sIdx : (N_S - 1);
      float sim = c[v] * invq * invS[sClamp];
      if (sIdx >= cEnd) sim = NEG_INF;
      if (sim > t4) { TOP5_INSERT(sim, sIdx); }
    }

    cur ^= 1;
  }

  // Publish per-lane top-5 and merge within the block.
  mv[lm][wave * 2 + lh][0] = t0; mi[lm][wave * 2 + lh][0] = i0;
  mv[lm][wave * 2 + lh][1] = t1; mi[lm][wave * 2 + lh][1] = i1;
  mv[lm][wave * 2 + lh][2] = t2; mi[lm][wave * 2 + lh][2] = i2;
  mv[lm][wave * 2 + lh][3] = t3; mi[lm][wave * 2 + lh][3] = i3;
  mv[lm][wave * 2 + lh][4] = t4; mi[lm][wave * 2 + lh][4] = i4;
  __syncthreads();

  if (tid < 16) {
    t0 = t1 = t2 = t3 = t4 = NEG_INF;
    i0 = i1 = i2 = i3 = i4 = 0;
#pragma unroll 1
    for (int L = 0; L < 2 * WAVES; ++L) {
#pragma unroll
      for (int j = 0; j < KNN; ++j) {
        float v = mv[tid][L][j];
        int   s = mi[tid][L][j];
        if (v > t4) { TOP5_INSERT(v, s); }
      }
    }
    const size_t o = ((size_t)(qbase + tid) * NSPLIT + split) * KNN;
    pVal[o + 0] = t0; pIdx[o + 0] = i0;
    pVal[o + 1] = t1; pIdx[o + 1] = i1;
    pVal[o + 2] = t2; pIdx[o + 2] = i2;
    pVal[o + 3] = t3; pIdx[o + 3] = i3;
    pVal[o + 4] = t4; pIdx[o + 4] = i4;
  }
}

// ---------------------------------------------------------------------------
// Kernel 3: merge 64 partial top-5s per query, compute weights, gather targets
// ---------------------------------------------------------------------------
__global__ void knn_finalize(const float* __restrict__ pVal,
                             const int* __restrict__ pIdx,
                             const float* __restrict__ T,
                             float* __restrict__ out) {
  const int q = blockIdx.x * blockDim.x + threadIdx.x;
  if (q >= B_Q) return;

  float t0 = NEG_INF, t1 = NEG_INF, t2 = NEG_INF, t3 = NEG_INF, t4 = NEG_INF;
  int   i0 = 0, i1 = 0, i2 = 0, i3 = 0, i4 = 0;
  const float* pv = pVal + (size_t)q * NSPLIT * KNN;
  const int*   pi = pIdx + (size_t)q * NSPLIT * KNN;
#pragma unroll 1
  for (int j = 0; j < NSPLIT * KNN; ++j) {
    float v = pv[j];
    int   s = pi[j];
    if (v > t4) { TOP5_INSERT(v, s); }
  }

  // distances of unit vectors: d = sqrt(max(2 - 2*sim, 0))
  float tv[KNN] = {t0, t1, t2, t3, t4};
  int   ti[KNN] = {i0, i1, i2, i3, i4};
  float w[KNN], wsum = 0.f;
#pragma unroll
  for (int j = 0; j < KNN; ++j) {
    float d = sqrtf(fmaxf(2.f - 2.f * tv[j], 0.f));
    w[j] = 1.f / (d + 1e-8f);
    wsum += w[j];
  }
  float acc[DT_];
#pragma unroll
  for (int t = 0; t < DT_; ++t) acc[t] = 0.f;
#pragma unroll
  for (int j = 0; j < KNN; ++j) {
    const float* trow = T + (size_t)ti[j] * DT_;
    const float wj = w[j] / wsum;
#pragma unroll
    for (int t = 0; t < DT_; ++t) acc[t] += wj * trow[t];
  }
#pragma unroll
  for (int t = 0; t < DT_; ++t) out[(size_t)q * DT_ + t] = acc[t];
}

// ---------------------------------------------------------------------------
extern "C" void kernel_launch(void* const* d_in, const int* in_sizes, int n_in,
                              void* d_out, int out_size, void* d_ws,
                              size_t ws_size, hipStream_t stream) {
  const float* Q = (const float*)d_in[0];
  const float* S = (const float*)d_in[1];
  const float* T = (const float*)d_in[2];
  // d_in[3] is k (==5), baked in as KNN.

  float* invQ = (float*)d_ws;                          // B_Q floats
  float* invS = invQ + B_Q;                            // N_S floats
  float* pVal = invS + N_S;                            // B_Q*NSPLIT*KNN floats
  int*   pIdx = (int*)(pVal + (size_t)B_Q * NSPLIT * KNN);
  float* out  = (float*)d_out;

  rownorm_inv<<<B_Q / 4, 128, 0, stream>>>(Q, invQ, B_Q);
  rownorm_inv<<<N_S / 4, 128, 0, stream>>>(S, invS, N_S);

  dim3 grid(B_Q / 16, NSPLIT);
  knn_gemm_topk<<<grid, 32 * WAVES, 0, stream>>>(Q, S, invQ, invS, pVal, pIdx);

  knn_finalize<<<(B_Q + 255) / 256, 256, 0, stream>>>(pVal, pIdx, T, out);
}